// GAT_4063039062672
// MI455X (gfx1250) — compile-verified
//
#include <hip/hip_runtime.h>
#include <hip/hip_bf16.h>
#include <math.h>

#define NNODES 30000
#define NEDGES 480000
#define FIN    100
#define HIDC   64
#define HEADS  4
#define NCLS   47
#define HC     256      // HEADS*HIDC
#define HC2    188      // HEADS*NCLS
#define NEG_SLOPE 0.2f
#define BN_EPS    1e-5f

typedef __attribute__((ext_vector_type(2))) float v2f;
typedef __attribute__((ext_vector_type(8))) float v8f;

// ---------------- fp32 WMMA GEMM: C[M,Nc] = A[M,K] @ B[K,Nc], row-major ----
// One wave computes a 16x64 strip (4 WMMA tiles), K-step 4 via
// v_wmma_f32_16x16x4_f32. blockDim.x = 256 (8 waves). grid.y = col strips.
// All fragment loads of a k-step are hoisted ahead of the 4 WMMAs so they
// issue as one clause with a single wait; B addressing is strength-reduced
// to a pointer bump. Tail strips are branch-free (clamp + 0/1 mask) so EXEC
// stays all-ones through the WMMA loop.
__global__ void gemm_wmma_f32(const float* __restrict__ A,
                              const float* __restrict__ B,
                              float* __restrict__ C,
                              int M, int K, int Nc) {
  const int wid  = threadIdx.x >> 5;
  const int lane = threadIdx.x & 31;
  const int rowTile = blockIdx.x * 8 + wid;
  const int row0 = rowTile * 16;
  if (row0 >= M) return;                 // wave-uniform: EXEC stays all-ones
  const int col0 = blockIdx.y * 64;
  const int l  = lane & 15;
  const int kk = (lane >> 4) << 1;       // 0 or 2 (K sub-pair per half-wave)

  const float*  Arow    = A + (size_t)(row0 + l) * K + kk;
  const size_t  bstride = (size_t)Nc * 4;   // 4 K-rows per step
  v8f acc[4] = {};

  if (col0 + 64 <= Nc) {
    // fast path: full strip in range, unconditional loads
    const float* Bp = B + (size_t)kk * Nc + col0 + l;
#pragma unroll 2
    for (int k0 = 0; k0 < K; k0 += 4) {
      v2f a; a.x = Arow[k0]; a.y = Arow[k0 + 1];
      v2f b0, b1, b2, b3;
      b0.x = Bp[0];  b0.y = Bp[Nc];
      b1.x = Bp[16]; b1.y = Bp[Nc + 16];
      b2.x = Bp[32]; b2.y = Bp[Nc + 32];
      b3.x = Bp[48]; b3.y = Bp[Nc + 48];
      acc[0] = __builtin_amdgcn_wmma_f32_16x16x4_f32(
          false, a, false, b0, (short)0, acc[0], false, false);
      acc[1] = __builtin_amdgcn_wmma_f32_16x16x4_f32(
          false, a, false, b1, (short)0, acc[1], false, false);
      acc[2] = __builtin_amdgcn_wmma_f32_16x16x4_f32(
          false, a, false, b2, (short)0, acc[2], false, false);
      acc[3] = __builtin_amdgcn_wmma_f32_16x16x4_f32(
          false, a, false, b3, (short)0, acc[3], false, false);
      Bp += bstride;
    }
  } else {
    // tail strip: clamp column address, zero via mask multiply (branch-free)
    int   bcol[4];
    float bmask[4];
#pragma unroll
    for (int t = 0; t < 4; ++t) {
      const int c = col0 + t * 16 + l;
      bmask[t] = (c < Nc) ? 1.0f : 0.0f;
      bcol[t]  = (c < Nc) ? c : (Nc - 1);
    }
    const float* Bp = B + (size_t)kk * Nc;
#pragma unroll 2
    for (int k0 = 0; k0 < K; k0 += 4) {
      v2f a; a.x = Arow[k0]; a.y = Arow[k0 + 1];
      v2f b0, b1, b2, b3;
      b0.x = Bp[bcol[0]] * bmask[0]; b0.y = Bp[Nc + bcol[0]] * bmask[0];
      b1.x = Bp[bcol[1]] * bmask[1]; b1.y = Bp[Nc + bcol[1]] * bmask[1];
      b2.x = Bp[bcol[2]] * bmask[2]; b2.y = Bp[Nc + bcol[2]] * bmask[2];
      b3.x = Bp[bcol[3]] * bmask[3]; b3.y = Bp[Nc + bcol[3]] * bmask[3];
      acc[0] = __builtin_amdgcn_wmma_f32_16x16x4_f32(
          false, a, false, b0, (short)0, acc[0], false, false);
      acc[1] = __builtin_amdgcn_wmma_f32_16x16x4_f32(
          false, a, false, b1, (short)0, acc[1], false, false);
      acc[2] = __builtin_amdgcn_wmma_f32_16x16x4_f32(
          false, a, false, b2, (short)0, acc[2], false, false);
      acc[3] = __builtin_amdgcn_wmma_f32_16x16x4_f32(
          false, a, false, b3, (short)0, acc[3], false, false);
      Bp += bstride;
    }
  }

  // C/D layout: VGPR r holds M = (lane>=16 ? 8 : 0)+r, N = lane&15
  const int mbase = row0 + ((lane >> 4) << 3);
#pragma unroll
  for (int t = 0; t < 4; ++t) {
    const int c = col0 + t * 16 + l;
    if (c < Nc) {
#pragma unroll
      for (int r = 0; r < 8; ++r)
        C[(size_t)(mbase + r) * Nc + c] = acc[t][r];
    }
  }
}

// ---------------- attention per-node dot products (wave per node) ----------
__global__ void attn_node(const float* __restrict__ h,
                          const float* __restrict__ a_src,
                          const float* __restrict__ a_dst,
                          float* __restrict__ als, float* __restrict__ ald,
                          int n_nodes, int C) {
  const int wave = (blockIdx.x * blockDim.x + threadIdx.x) >> 5;
  const int lane = threadIdx.x & 31;
  if (wave >= n_nodes) return;
  const float* hr = h + (size_t)wave * HEADS * C;
#pragma unroll
  for (int hd = 0; hd < HEADS; ++hd) {
    float vs = 0.f, vd = 0.f;
    for (int c = lane; c < C; c += 32) {
      const float hv = hr[hd * C + c];
      vs += hv * a_src[hd * C + c];
      vd += hv * a_dst[hd * C + c];
    }
    for (int o = 16; o; o >>= 1) {
      vs += __shfl_xor(vs, o, 32);
      vd += __shfl_xor(vd, o, 32);
    }
    if (lane == 0) {
      als[wave * HEADS + hd] = vs;
      ald[wave * HEADS + hd] = vd;
    }
  }
}

// monotone float<->u32 encoding so integer atomicMax orders floats correctly
__device__ __forceinline__ unsigned enc_f(float f) {
  unsigned u = __float_as_uint(f);
  return (u & 0x80000000u) ? ~u : (u | 0x80000000u);
}
__device__ __forceinline__ float dec_f(unsigned u) {
  return (u & 0x80000000u) ? __uint_as_float(u & 0x7FFFFFFFu)
                           : __uint_as_float(~u);
}

__global__ void fill_u32(unsigned* __restrict__ p, unsigned v, int n) {
  const int i = blockIdx.x * blockDim.x + threadIdx.x;
  if (i < n) p[i] = v;
}

// pass 1: leaky-relu score + segment max (encoded atomicMax)
__global__ void edge_score_max(const float* __restrict__ als,
                               const float* __restrict__ ald,
                               const int* __restrict__ src,
                               const int* __restrict__ dst,
                               float* __restrict__ es,
                               unsigned* __restrict__ menc, int n_edges) {
  const int idx = blockIdx.x * blockDim.x + threadIdx.x;
  if (idx >= n_edges * HEADS) return;
  const int e = idx >> 2, hd = idx & 3;
  const int s = src[e], d = dst[e];
  float sc = als[s * HEADS + hd] + ald[d * HEADS + hd];
  sc = sc > 0.f ? sc : NEG_SLOPE * sc;
  es[idx] = sc;
  atomicMax(&menc[d * HEADS + hd], enc_f(sc));
}

// pass 2: exp(score - max) + segment sum (denominator)
__global__ void edge_exp(float* __restrict__ es,
                         const unsigned* __restrict__ menc,
                         float* __restrict__ den,
                         const int* __restrict__ dst, int n_edges) {
  const int idx = blockIdx.x * blockDim.x + threadIdx.x;
  if (idx >= n_edges * HEADS) return;
  const int e = idx >> 2, hd = idx & 3;
  const int d = dst[e];
  const float w = expf(es[idx] - dec_f(menc[d * HEADS + hd]));
  es[idx] = w;
  atomicAdd(&den[d * HEADS + hd], w);
}

// pass 3: scatter alpha-weighted messages (wave per edge, strided channels)
__global__ void edge_agg(const float* __restrict__ h,
                         const float* __restrict__ es,
                         const float* __restrict__ den,
                         const int* __restrict__ src,
                         const int* __restrict__ dst,
                         float* __restrict__ agg,
                         int n_edges, int HCt, int C) {
  const int wave = (blockIdx.x * blockDim.x + threadIdx.x) >> 5;
  const int lane = threadIdx.x & 31;
  if (wave >= n_edges) return;
  const int s = src[wave], d = dst[wave];
  float alpha[HEADS];
#pragma unroll
  for (int hd = 0; hd < HEADS; ++hd)
    alpha[hd] = es[wave * HEADS + hd] / (den[d * HEADS + hd] + 1e-16f);
  const float* hs = h + (size_t)s * HCt;
  float* ad = agg + (size_t)d * HCt;
  for (int c = lane; c < HCt; c += 32)
    atomicAdd(&ad[c], hs[c] * alpha[c / C]);
}

// t = agg + bias + skip + sbias (in place into agg) + per-channel sum/sumsq
__global__ void finalize_stats(float* __restrict__ t,
                               const float* __restrict__ skip,
                               const float* __restrict__ bias,
                               const float* __restrict__ sbias,
                               float* __restrict__ stats, int n_nodes) {
  const int c = threadIdx.x;               // blockDim.x == HC == 256
  const int row0 = blockIdx.x * 128;
  const int rowEnd = min(row0 + 128, n_nodes);
  const float bsum = bias[c] + sbias[c];
  float s1 = 0.f, s2 = 0.f;
  for (int n = row0; n < rowEnd; ++n) {
    const size_t i = (size_t)n * HC + c;
    const float v = t[i] + skip[i] + bsum;
    t[i] = v;
    s1 += v; s2 += v * v;
  }
  atomicAdd(&stats[c], s1);
  atomicAdd(&stats[HC + c], s2);
}

__global__ void bn_elu(const float* __restrict__ t,
                       const float* __restrict__ stats,
                       const float* __restrict__ g,
                       const float* __restrict__ be,
                       float* __restrict__ out, int total) {
  const int i = blockIdx.x * blockDim.x + threadIdx.x;
  if (i >= total) return;
  const int c = i & (HC - 1);
  const float inv_n = 1.0f / (float)NNODES;
  const float mu  = stats[c] * inv_n;
  const float var = stats[HC + c] * inv_n - mu * mu;
  const float y = g[c] * (t[i] - mu) * rsqrtf(var + BN_EPS) + be[c];
  out[i] = y > 0.f ? y : expm1f(y);
}

// layer 2: head-mean + bias + skip + log_softmax (wave per node, 47 classes)
__global__ void final2(const float* __restrict__ agg,
                       const float* __restrict__ skip,
                       const float* __restrict__ b2,
                       const float* __restrict__ sb2,
                       float* __restrict__ out, int n_nodes) {
  const int wave = (blockIdx.x * blockDim.x + threadIdx.x) >> 5;
  const int lane = threadIdx.x & 31;
  if (wave >= n_nodes) return;
  const float* ag = agg + (size_t)wave * HC2;
  const float* sk = skip + (size_t)wave * NCLS;
  float v[2];
#pragma unroll
  for (int j = 0; j < 2; ++j) {
    const int c = lane + 32 * j;
    v[j] = -3.402823466e38f;
    if (c < NCLS)
      v[j] = (ag[c] + ag[NCLS + c] + ag[2 * NCLS + c] + ag[3 * NCLS + c]) * 0.25f
             + b2[c] + sk[c] + sb2[c];
  }
  float mx = fmaxf(v[0], v[1]);
  for (int o = 16; o; o >>= 1) mx = fmaxf(mx, __shfl_xor(mx, o, 32));
  float se = 0.f;
#pragma unroll
  for (int j = 0; j < 2; ++j) {
    const int c = lane + 32 * j;
    if (c < NCLS) se += expf(v[j] - mx);
  }
  for (int o = 16; o; o >>= 1) se += __shfl_xor(se, o, 32);
  const float lse = mx + logf(se);
#pragma unroll
  for (int j = 0; j < 2; ++j) {
    const int c = lane + 32 * j;
    if (c < NCLS) out[(size_t)wave * NCLS + c] = v[j] - lse;
  }
}

extern "C" void kernel_launch(void* const* d_in, const int* in_sizes, int n_in,
                              void* d_out, int out_size, void* d_ws, size_t ws_size,
                              hipStream_t stream) {
  const float* x   = (const float*)d_in[0];
  const int*   ei  = (const int*)d_in[1];
  const int*   src = ei;
  const int*   dst = ei + NEDGES;
  const float* w0  = (const float*)d_in[2];
  const float* as0 = (const float*)d_in[3];
  const float* ad0 = (const float*)d_in[4];
  const float* b0  = (const float*)d_in[5];
  const float* sw0 = (const float*)d_in[6];
  const float* sb0 = (const float*)d_in[7];
  const float* g0  = (const float*)d_in[8];
  const float* be0 = (const float*)d_in[9];
  const float* w1  = (const float*)d_in[10];
  const float* as1 = (const float*)d_in[11];
  const float* ad1 = (const float*)d_in[12];
  const float* b1  = (const float*)d_in[13];
  const float* sw1 = (const float*)d_in[14];
  const float* sb1 = (const float*)d_in[15];
  const float* g1  = (const float*)d_in[16];
  const float* be1 = (const float*)d_in[17];
  const float* w2  = (const float*)d_in[18];
  const float* as2 = (const float*)d_in[19];
  const float* ad2 = (const float*)d_in[20];
  const float* b2  = (const float*)d_in[21];
  const float* sw2 = (const float*)d_in[22];
  const float* sb2 = (const float*)d_in[23];

  // workspace layout (floats)
  float* ws = (float*)d_ws;
  size_t off = 0;
  float* H   = ws + off; off += (size_t)NNODES * HC;   // GAT linear output
  float* S   = ws + off; off += (size_t)NNODES * HC;   // skip linear output
  float* AGG = ws + off; off += (size_t)NNODES * HC;   // aggregation / t
  float* A1  = ws + off; off += (size_t)NNODES * HC;   // activated layer input
  float* ALS = ws + off; off += (size_t)NNODES * HEADS;
  float* ALD = ws + off; off += (size_t)NNODES * HEADS;
  float* ES  = ws + off; off += (size_t)NEDGES * HEADS;
  unsigned* MEN = (unsigned*)(ws + off); off += (size_t)NNODES * HEADS;
  float* DEN = ws + off; off += (size_t)NNODES * HEADS;
  float* ST  = ws + off; off += 2 * HC;

  const dim3 blk(256);
  const int gemm_gx   = (NNODES / 16 + 7) / 8;                 // 235
  const int attn_gx   = (NNODES * 32 + 255) / 256;             // 3750
  const int edge_gx   = (NEDGES * HEADS + 255) / 256;          // 7500
  const int eagg_gx   = (NEDGES * 32 + 255) / 256;             // 60000
  const int fs_gx     = (NNODES + 127) / 128;                  // 235
  const int bn_gx     = (NNODES * HC + 255) / 256;             // 30000
  const int fillNH_gx = (NNODES * HEADS + 255) / 256;
  const int fillHC_gx = (NNODES * HC + 255) / 256;
  const int fillH2_gx = (NNODES * HC2 + 255) / 256;

  // ---------------- layer 0 ----------------
  gemm_wmma_f32<<<dim3(gemm_gx, 4), blk, 0, stream>>>(x, w0,  H, NNODES, FIN, HC);
  gemm_wmma_f32<<<dim3(gemm_gx, 4), blk, 0, stream>>>(x, sw0, S, NNODES, FIN, HC);
  attn_node<<<attn_gx, blk, 0, stream>>>(H, as0, ad0, ALS, ALD, NNODES, HIDC);
  fill_u32<<<fillNH_gx, blk, 0, stream>>>(MEN, 0u, NNODES * HEADS);
  fill_u32<<<fillNH_gx, blk, 0, stream>>>((unsigned*)DEN, 0u, NNODES * HEADS);
  fill_u32<<<fillHC_gx, blk, 0, stream>>>((unsigned*)AGG, 0u, NNODES * HC);
  fill_u32<<<2, blk, 0, stream>>>((unsigned*)ST, 0u, 2 * HC);
  edge_score_max<<<edge_gx, blk, 0, stream>>>(ALS, ALD, src, dst, ES, MEN, NEDGES);
  edge_exp<<<edge_gx, blk, 0, stream>>>(ES, MEN, DEN, dst, NEDGES);
  edge_agg<<<eagg_gx, blk, 0, stream>>>(H, ES, DEN, src, dst, AGG, NEDGES, HC, HIDC);
  finalize_stats<<<fs_gx, blk, 0, stream>>>(AGG, S, b0, sb0, ST, NNODES);
  bn_elu<<<bn_gx, blk, 0, stream>>>(AGG, ST, g0, be0, A1, NNODES * HC);

  // ---------------- layer 1 ----------------
  gemm_wmma_f32<<<dim3(gemm_gx, 4), blk, 0, stream>>>(A1, w1,  H, NNODES, HC, HC);
  gemm_wmma_f32<<<dim3(gemm_gx, 4), blk, 0, stream>>>(A1, sw1, S, NNODES, HC, HC);
  attn_node<<<attn_gx, blk, 0, stream>>>(H, as1, ad1, ALS, ALD, NNODES, HIDC);
  fill_u32<<<fillNH_gx, blk, 0, stream>>>(MEN, 0u, NNODES * HEADS);
  fill_u32<<<fillNH_gx, blk, 0, stream>>>((unsigned*)DEN, 0u, NNODES * HEADS);
  fill_u32<<<fillHC_gx, blk, 0, stream>>>((unsigned*)AGG, 0u, NNODES * HC);
  fill_u32<<<2, blk, 0, stream>>>((unsigned*)ST, 0u, 2 * HC);
  edge_score_max<<<edge_gx, blk, 0, stream>>>(ALS, ALD, src, dst, ES, MEN, NEDGES);
  edge_exp<<<edge_gx, blk, 0, stream>>>(ES, MEN, DEN, dst, NEDGES);
  edge_agg<<<eagg_gx, blk, 0, stream>>>(H, ES, DEN, src, dst, AGG, NEDGES, HC, HIDC);
  finalize_stats<<<fs_gx, blk, 0, stream>>>(AGG, S, b1, sb1, ST, NNODES);
  bn_elu<<<bn_gx, blk, 0, stream>>>(AGG, ST, g1, be1, A1, NNODES * HC);

  // ---------------- layer 2 (concat=False, mean over heads) ----------------
  gemm_wmma_f32<<<dim3(gemm_gx, 3), blk, 0, stream>>>(A1, w2,  H, NNODES, HC, HC2);
  gemm_wmma_f32<<<dim3(gemm_gx, 1), blk, 0, stream>>>(A1, sw2, S, NNODES, HC, NCLS);
  attn_node<<<attn_gx, blk, 0, stream>>>(H, as2, ad2, ALS, ALD, NNODES, NCLS);
  fill_u32<<<fillNH_gx, blk, 0, stream>>>(MEN, 0u, NNODES * HEADS);
  fill_u32<<<fillNH_gx, blk, 0, stream>>>((unsigned*)DEN, 0u, NNODES * HEADS);
  fill_u32<<<fillH2_gx, blk, 0, stream>>>((unsigned*)AGG, 0u, NNODES * HC2);
  edge_score_max<<<edge_gx, blk, 0, stream>>>(ALS, ALD, src, dst, ES, MEN, NEDGES);
  edge_exp<<<edge_gx, blk, 0, stream>>>(ES, MEN, DEN, dst, NEDGES);
  edge_agg<<<eagg_gx, blk, 0, stream>>>(H, ES, DEN, src, dst, AGG, NEDGES, HC2, NCLS);
  final2<<<attn_gx, blk, 0, stream>>>(AGG, S, b2, sb2, (float*)d_out, NNODES);
}